// ConvNetFullOpti_57372173140201
// MI455X (gfx1250) — compile-verified
//
#include <hip/hip_runtime.h>
#include <math.h>

// ---------------------------------------------------------------------------
// ConvNet + Markowitz QP head, MI455X (gfx1250, wave32, WMMA).
//
//   A) BatchNorm batch stats (sum / sumsq -> scale/shift)         [bandwidth]
//   A2) bpack: weights repacked into WMMA lane order, masks folded [once, L2]
//   B) fused: BN + conv(3x1,s2) -> LDS, WMMA f32 16x16x4 GEMMs
//      (lin: 120->5 tanh, linc: 120->100 relu), 5x5 covariance    [WMMA]
//      NOTE: eigh+sqrt deleted: Q = V clip(l,1e-8) V^T == cov to <1e-8 abs.
//   C) QP: per-lane PGD (150 iters), projection onto {sum=1,|w|<=1}
//      via 30-iter bisection (fp32-exact) + active-set correction [VALU]
// ---------------------------------------------------------------------------

typedef __attribute__((ext_vector_type(2))) float v2f;
typedef __attribute__((ext_vector_type(8))) float v8f;

#define NFEAT    250
#define NCONV    120          // 24*5 conv outputs per channel
#define FSTR     124          // padded LDS stride (bank-conflict free)
#define HSTR     104          // padded h stride
#define PGD_ITERS 150
#define BIS_ITERS 30

// ws float layout:
//  [0..249] sum  [250..499] sumsq  [500..749] scale  [750..999] shift
//  [1024 .. 16384)      bpack: 8 tiles x 30 steps x 32 lanes x 2 f32 (15360 f)
//  [16384 .. +25N)      Q (N x 5 x 5)
//  [16384+25N .. +5N)   rets (N x 5)
#define BPACK_OFF 1024
#define COVQ_OFF  16384

// ------------------------------ BN stats -----------------------------------

__global__ __launch_bounds__(256) void bn_zero_kernel(float* ws) {
  int t = threadIdx.x;
  if (t < NFEAT) { ws[t] = 0.f; ws[NFEAT + t] = 0.f; }
}

__global__ __launch_bounds__(256) void bn_stats_kernel(const float* __restrict__ x,
                                                       float* __restrict__ ws, int N) {
  int f = threadIdx.x;
  if (f >= NFEAT) return;
  int s0 = blockIdx.x * 256;
  int cnt = (N - s0 < 256) ? (N - s0) : 256;
  float s = 0.f, sq = 0.f;
  for (int i = 0; i < cnt; ++i) {
    float v = x[(size_t)(s0 + i) * NFEAT + f];   // 250 lanes read one contiguous row
    s += v;
    sq = fmaf(v, v, sq);
  }
  atomicAdd(&ws[f], s);
  atomicAdd(&ws[NFEAT + f], sq);
}

__global__ __launch_bounds__(256) void bn_final_kernel(float* __restrict__ ws,
                                                       const float* __restrict__ gamma,
                                                       const float* __restrict__ beta, int N) {
  int f = threadIdx.x;
  if (f >= NFEAT) return;
  float invN = 1.0f / (float)N;
  float mu   = ws[f] * invN;
  float var  = ws[NFEAT + f] * invN - mu * mu;     // biased var, as reference
  float rs   = rsqrtf(var + 1e-5f);
  float sc   = gamma[f] * rs;
  ws[500 + f] = sc;
  ws[750 + f] = beta[f] - mu * sc;
}

// --------------------- WMMA B-fragment weight packing ----------------------
// bpack[((T*30 + t)*32 + lane)*2 + {0,1}]:
//   tiles 0..6 -> linc_w rows T*16 + (lane&15) (0 beyond row 99)
//   tile  7    -> lin_w  rows (lane&15)        (0 beyond row 4)
//   k = 4*t + 2*(lane>>4)   (16x16x4 f32 B-fragment K offset)

__global__ __launch_bounds__(256) void bpack_kernel(const float* __restrict__ lin_w,
                                                    const float* __restrict__ linc_w,
                                                    float* __restrict__ bpack) {
  int e = blockIdx.x * 256 + threadIdx.x;      // 0..7679
  if (e >= 8 * 30 * 32) return;
  int l    = e & 31;
  int rest = e >> 5;                           // T*30 + t
  int T    = rest / 30;
  int t    = rest - T * 30;
  int k    = 4 * t + ((l >> 4) << 1);
  int n16  = l & 15;
  float bx = 0.f, by = 0.f;
  if (T == 7) {
    if (n16 < 5) { bx = lin_w[n16 * 120 + k]; by = lin_w[n16 * 120 + k + 1]; }
  } else {
    int n = T * 16 + n16;
    if (n < 100) { bx = linc_w[n * 120 + k]; by = linc_w[n * 120 + k + 1]; }
  }
  bpack[e * 2]     = bx;
  bpack[e * 2 + 1] = by;
}

// --------------------------- fused WMMA kernel -----------------------------
// block = 128 threads = 4 waves; each wave owns 16 consecutive samples.

__global__ __launch_bounds__(128) void fused_kernel(
    const float* __restrict__ x,
    const float* __restrict__ ws,          // scale at +500, shift at +750
    const float* __restrict__ conv_w,      // 6 floats: [o][k]
    const float* __restrict__ conv_b,      // 2
    const float* __restrict__ lin_b,       // 5
    const float* __restrict__ linc_b,      // 100
    const float* __restrict__ bpack,       // packed B fragments
    float* __restrict__ covQ,              // N x 25
    float* __restrict__ rets_out,          // N x 5
    int N) {
  __shared__ float smem[4 * 2 * 16 * FSTR];          // 63488 B

  const int tid  = threadIdx.x;
  const int wave = tid >> 5;
  const int lane = tid & 31;
  const int l16  = lane & 15;
  const int hi   = lane >> 4;          // half-wave select
  const int koff = hi * 2;             // K-offset inside a 4-wide step
  const int base = blockIdx.x * 64 + wave * 16;      // first sample of this wave
  if (blockIdx.x * 64 >= N) return;

  const float* scale = ws + 500;
  const float* shift = ws + 750;
  float* featw = &smem[wave * (2 * 16 * FSTR)];      // [ch*16*FSTR + s*FSTR + j]

  // keep the 60KB packed-weight panel warm in L2 (global_prefetch_b8 path)
  if (tid < 15) __builtin_prefetch(bpack + tid * 1024, 0, 1);

  const float cw0 = conv_w[0], cw1 = conv_w[1], cw2 = conv_w[2];
  const float cw3 = conv_w[3], cw4 = conv_w[4], cw5 = conv_w[5];
  const float cb0 = conv_b[0], cb1 = conv_b[1];

  // ---- phase 1: BN + conv into LDS (16 samples x 120 x 2ch per wave) ----
  for (int it = 0; it < 60; ++it) {
    int p = it * 32 + lane;            // 0..1919
    int s = p / NCONV;
    int j = p - s * NCONV;             // conv position h*5+a
    int h = j / 5;
    int a = j - h * 5;
    int f0 = 10 * h + a;               // input row 2h, col a
    const float* xs = x + (size_t)(base + s) * NFEAT;
    float x0 = fmaf(xs[f0],      scale[f0],      shift[f0]);
    float x1 = fmaf(xs[f0 + 5],  scale[f0 + 5],  shift[f0 + 5]);
    float x2 = fmaf(xs[f0 + 10], scale[f0 + 10], shift[f0 + 10]);
    float c0 = fmaf(x2, cw2, fmaf(x1, cw1, fmaf(x0, cw0, cb0)));
    float c1 = fmaf(x2, cw5, fmaf(x1, cw4, fmaf(x0, cw3, cb1)));
    featw[s * FSTR + j]               = c0;   // ch0: x_exp
    featw[16 * FSTR + s * FSTR + j]   = c1;   // ch1: x_cov
  }
  __syncthreads();

  // ---- phase 2: WMMA GEMMs (f32 16x16x4, K=120 in 30 steps) ----
  // rets head (packed tile 7): 16 samples x 5 outputs
  v8f cr = {0.f, 0.f, 0.f, 0.f, 0.f, 0.f, 0.f, 0.f};
  {
    const float* bpr = bpack + (7 * 30) * 64 + lane * 2;   // 64 f32 per step
#pragma unroll
    for (int t = 0; t < 30; ++t) {
      int k = 4 * t + koff;
      v2f a;
      a.x = featw[l16 * FSTR + k];
      a.y = featw[l16 * FSTR + k + 1];
      v2f b = *reinterpret_cast<const v2f*>(bpr + t * 64);  // coalesced
      cr = __builtin_amdgcn_wmma_f32_16x16x4_f32(false, a, false, b,
                                                 (short)0, cr, false, false);
    }
  }

  // covariance head: 16 samples x 100 outputs (7 column tiles)
  v8f ch[7];
#pragma unroll
  for (int tile = 0; tile < 7; ++tile) {
    v8f acc = {0.f, 0.f, 0.f, 0.f, 0.f, 0.f, 0.f, 0.f};
    const float* bpt = bpack + (tile * 30) * 64 + lane * 2;
#pragma unroll
    for (int t = 0; t < 30; ++t) {
      int k = 4 * t + koff;
      v2f a;
      a.x = featw[16 * FSTR + l16 * FSTR + k];
      a.y = featw[16 * FSTR + l16 * FSTR + k + 1];
      v2f b = *reinterpret_cast<const v2f*>(bpt + t * 64);  // coalesced
      acc = __builtin_amdgcn_wmma_f32_16x16x4_f32(false, a, false, b,
                                                  (short)0, acc, false, false);
    }
    ch[tile] = acc;
  }

  // ---- epilogue: tanh(rets) -> global; relu(h) -> LDS ----
  if (l16 < 5) {
    float bb = lin_b[l16];
#pragma unroll
    for (int r = 0; r < 8; ++r) {
      int m = r + hi * 8;
      rets_out[(size_t)(base + m) * 5 + l16] = tanhf(cr[r] + bb);
    }
  }
  __syncthreads();                         // done reading featw
  float* hW = featw;                       // reuse per-wave region: [s*HSTR + out]
#pragma unroll
  for (int tile = 0; tile < 7; ++tile) {
    int n = tile * 16 + l16;
    if (n < 100) {
      float bb = linc_b[n];
#pragma unroll
      for (int r = 0; r < 8; ++r) {
        int m = r + hi * 8;
        hW[m * HSTR + n] = fmaxf(ch[tile][r] + bb, 0.f);
      }
    }
  }
  __syncthreads();

  // ---- phase 3: cov[i][j] = (sum h_i h_j - 20 m_i m_j)/19, h is (20,5) ----
  for (int it = 0; it < 8; ++it) {
    int e = it * 32 + lane;                // 16 samples * 15 pairs = 240
    if (e >= 240) break;
    int s = e / 15;
    int p = e - s * 15;
    int i = (p < 5) ? 0 : (p < 9) ? 1 : (p < 12) ? 2 : (p < 14) ? 3 : 4;
    int pb = (i == 0) ? 0 : (i == 1) ? 5 : (i == 2) ? 9 : (i == 3) ? 12 : 14;
    int j = i + (p - pb);
    const float* hs = hW + s * HSTR;
    float si = 0.f, sj = 0.f, sij = 0.f;
#pragma unroll
    for (int t = 0; t < 20; ++t) {
      float ai = hs[t * 5 + i];
      float aj = hs[t * 5 + j];
      si += ai;
      sj += aj;
      sij = fmaf(ai, aj, sij);
    }
    float c = (sij - si * sj * (1.f / 20.f)) * (1.f / 19.f);
    size_t o = (size_t)(base + s) * 25;
    covQ[o + i * 5 + j] = c;
    if (i != j) covQ[o + j * 5 + i] = c;
  }
}

// ------------------------------- QP kernel ---------------------------------
// one lane per sample; everything in registers.

__global__ __launch_bounds__(256) void qp_kernel(const float* __restrict__ covQ,
                                                 const float* __restrict__ rets,
                                                 float* __restrict__ out, int N) {
  int s = blockIdx.x * blockDim.x + threadIdx.x;
  if (s >= N) return;

  float Q[25], r[5], w[5];
#pragma unroll
  for (int i = 0; i < 25; ++i) Q[i] = covQ[(size_t)s * 25 + i];
#pragma unroll
  for (int i = 0; i < 5; ++i) r[i] = rets[(size_t)s * 5 + i];
#pragma unroll
  for (int i = 0; i < 5; ++i) w[i] = 0.2f;

  float trQ = Q[0] + Q[6] + Q[12] + Q[18] + Q[24];
  float eta = 0.5f / (trQ + 1e-6f);

  for (int it = 0; it < PGD_ITERS; ++it) {
    float v[5];
#pragma unroll
    for (int i = 0; i < 5; ++i) {
      float g = 0.f;
#pragma unroll
      for (int j = 0; j < 5; ++j) g = fmaf(Q[i * 5 + j], w[j], g);
      v[i] = w[i] - eta * (2.f * g - r[i]);
    }
    // projection onto {sum w = 1, -1 <= w <= 1}
    float vmin = v[0], vmax = v[0];
#pragma unroll
    for (int i = 1; i < 5; ++i) { vmin = fminf(vmin, v[i]); vmax = fmaxf(vmax, v[i]); }
    float lo = vmin - 2.f, hh = vmax + 2.f;
    for (int b = 0; b < BIS_ITERS; ++b) {
      float mid = 0.5f * (lo + hh);
      float ssum = 0.f;
#pragma unroll
      for (int i = 0; i < 5; ++i)
        ssum += fminf(fmaxf(v[i] - mid, -1.f), 1.f);
      bool gt = ssum > 1.f;
      lo = gt ? mid : lo;
      hh = gt ? hh : mid;
    }
    float tau = 0.5f * (lo + hh);
    float w0s = 0.f, na = 0.f;
    float w0[5], act[5];
#pragma unroll
    for (int i = 0; i < 5; ++i) {
      float d = v[i] - tau;
      w0[i]  = fminf(fmaxf(d, -1.f), 1.f);
      act[i] = (fabsf(d) < 1.f - 1e-7f) ? 1.f : 0.f;
      w0s += w0[i];
      na  += act[i];
    }
    na = fmaxf(na, 1.f);
    float corr = (1.f - w0s) / na;
#pragma unroll
    for (int i = 0; i < 5; ++i) w[i] = w0[i] + act[i] * corr;
  }
#pragma unroll
  for (int i = 0; i < 5; ++i) out[(size_t)s * 5 + i] = w[i];
}

// ------------------------------- launcher ----------------------------------

extern "C" void kernel_launch(void* const* d_in, const int* in_sizes, int n_in,
                              void* d_out, int out_size, void* d_ws, size_t ws_size,
                              hipStream_t stream) {
  const float* x       = (const float*)d_in[0];
  const float* gamma   = (const float*)d_in[1];
  const float* beta    = (const float*)d_in[2];
  const float* conv_w  = (const float*)d_in[3];
  const float* conv_b  = (const float*)d_in[4];
  const float* lin_w   = (const float*)d_in[5];
  const float* lin_b   = (const float*)d_in[6];
  const float* linc_w  = (const float*)d_in[7];
  const float* linc_b  = (const float*)d_in[8];
  float* out = (float*)d_out;
  float* ws  = (float*)d_ws;

  const int N = in_sizes[0] / NFEAT;      // 65536
  float* bpack = ws + BPACK_OFF;
  float* covQ  = ws + COVQ_OFF;
  float* rets  = ws + COVQ_OFF + (size_t)25 * N;

  bn_zero_kernel<<<1, 256, 0, stream>>>(ws);
  bn_stats_kernel<<<(N + 255) / 256, 256, 0, stream>>>(x, ws, N);
  bn_final_kernel<<<1, 256, 0, stream>>>(ws, gamma, beta, N);
  bpack_kernel<<<30, 256, 0, stream>>>(lin_w, linc_w, bpack);
  fused_kernel<<<N / 64, 128, 0, stream>>>(x, ws, conv_w, conv_b, lin_b,
                                           linc_b, bpack, covQ, rets, N);
  qp_kernel<<<(N + 255) / 256, 256, 0, stream>>>(covQ, rets, out, N);
}